// TranslationalVariantConv2D_5798205849933
// MI455X (gfx1250) — compile-verified
//
#include <hip/hip_runtime.h>

typedef float v2f __attribute__((ext_vector_type(2)));
typedef float v8f __attribute__((ext_vector_type(8)));
typedef unsigned int v4u __attribute__((ext_vector_type(4)));
typedef int v4i __attribute__((ext_vector_type(4)));
typedef int v8i __attribute__((ext_vector_type(8)));

#define BT      64            // batches per workgroup (Btotal assumed multiple of BT)
#define THREADS 128           // 4 waves; each wave owns 16 batches
#define XT      (BT * 196)    // 12544 floats staged per workgroup

__global__ __launch_bounds__(THREADS)
void lc_conv2d_wmma_f32(const float* __restrict__ x,
                        const float* __restrict__ w,
                        float* __restrict__ out,
                        int Btotal) {
    // x tile: BT batches x 196 floats (14x14), linear layout          (50176 B)
    __shared__ __align__(16) float sx[XT];
    // pre-masked block-diagonal B fragments: [i][jg*4+jj][lane][2]    (14336 B)
    __shared__ __align__(16) float swb[7 * 8 * 64];

    const int tid  = threadIdx.x;
    const int lane = tid & 31;
    const int wave = tid >> 5;

    const long long xbase = (long long)blockIdx.x * XT;   // element offset into x

    // ---- stage x tile: Tensor Data Mover (async DMA global -> LDS) ----
#if __has_builtin(__builtin_amdgcn_tensor_load_to_lds) && __has_builtin(__builtin_amdgcn_s_wait_tensorcnt)
    if (wave == 0) {
        long long remain = (long long)Btotal * 196 - xbase;
        unsigned int td0 = (remain >= (long long)XT) ? (unsigned)XT
                         : (remain > 0 ? (unsigned)remain : 1u);
        unsigned long long gaddr  = (unsigned long long)(const void*)(x + xbase);
        unsigned int       ldsoff = (unsigned int)(unsigned long long)(void*)&sx[0];

        v4u g0;
        g0[0] = 1u;                                            // count=1, user desc
        g0[1] = ldsoff;                                        // lds_addr
        g0[2] = (unsigned int)(gaddr & 0xFFFFFFFFu);           // global_addr[31:0]
        g0[3] = (unsigned int)((gaddr >> 32) & 0x01FFFFFFu)    // global_addr[56:32]
              | (2u << 30);                                    // type = 2 (image)
        v8i g1;
        g1[0] = (int)(2u << 16);                               // data_size = 4B, no multicast
        g1[1] = (int)((td0 & 0xFFFFu) << 16);                  // tensor_dim0[15:0]
        g1[2] = (int)(((td0 >> 16) & 0xFFFFu) | (1u << 16));   // tensor_dim0[31:16] | tensor_dim1=1
        g1[3] = (int)((unsigned)(XT & 0xFFFF) << 16);          // tile_dim0 = 12544
        g1[4] = 0;                                             // tile_dim1=0, tile_dim2=0
        g1[5] = (int)XT;                                       // tensor_dim0_stride
        g1[6] = 0;
        g1[7] = 0;
        v4i g2 = {0, 0, 0, 0};
        v4i g3 = {0, 0, 0, 0};
        v8i g4 = {0, 0, 0, 0, 0, 0, 0, 0};
        __builtin_amdgcn_tensor_load_to_lds(g0, g1, g2, g3, g4, 0);
        __builtin_amdgcn_s_wait_tensorcnt(0);
    }
#else
    {
        const float4* xg  = (const float4*)x;
        float4*       sx4 = (float4*)sx;
        const int base = blockIdx.x * (BT * 49);
        for (int idx = tid; idx < BT * 49; idx += THREADS) {
            sx4[idx] = xg[base + idx];
        }
    }
#endif

    // ---- build pre-masked B fragments (overlaps with TDM) ----
    // B(4x16) for (i, jg, jj): B[k][n] = (n>>2 == jj) ? W[i, jg*4+jj, k, n&3] : 0
    // lane layout: lanes 0-15 hold K={0,1}, lanes 16-31 hold K={2,3}
    for (int idx = tid; idx < 7 * 8 * 32; idx += THREADS) {
        const int l    = idx & 31;
        const int slot = idx >> 5;          // i*8 + jg*4 + jj
        const int i    = slot >> 3;
        const int j    = slot & 7;          // jg*4 + jj
        const int jj   = j & 3;
        const int n    = l & 15;
        const int kb   = (l >> 4) * 2;
        const int c    = n & 3;
        float2 v = {0.f, 0.f};
        if (j < 7 && (n >> 2) == jj) {
            const float* wp = w + (i * 7 + j) * 16;   // (k,c) row-major, 4x4
            v.x = wp[kb * 4 + c];
            v.y = wp[(kb + 1) * 4 + c];
        }
        ((float2*)swb)[idx] = v;
    }
    __syncthreads();

    const int m    = lane & 15;   // A row M (batch); B/D column N
    const int half = lane >> 4;   // K pair select / D row-half select

    const long long b0 = (long long)blockIdx.x * BT + wave * 16;

    for (int i = 0; i < 7; ++i) {
        const int xrow = (wave * 16 + m) * 196 + (2 * i + half) * 14;
        for (int jg = 0; jg < 2; ++jg) {
            const int j0  = jg * 4;
            const int njj = 4 - jg;       // j = 0..6 (skip j=7)
            v8f acc = {0.f, 0.f, 0.f, 0.f, 0.f, 0.f, 0.f, 0.f};

            for (int jj = 0; jj < njj; ++jj) {
                const v2f a  = *(const v2f*)&sx[xrow + 2 * (j0 + jj)];
                const v2f bm = *(const v2f*)&swb[((i * 8 + j0 + jj) * 32 + lane) * 2];
                acc = __builtin_amdgcn_wmma_f32_16x16x4_f32(
                    false, a, false, bm, (short)0, acc, false, false);
            }

            // D tile: lane column n=m covers (j_local=n>>2, c=n&3); rows = 8 batches
            const int  colg   = i * 28 + j0 * 4 + m;
            const bool jvalid = (j0 + (m >> 2)) < 7;   // single lane predicate for all 8 stores
            float* op = out + (b0 + half * 8) * 196 + colg;
            if (jvalid) {
                #pragma unroll
                for (int r = 0; r < 8; ++r) op[r * 196] = acc[r];
            }
        }
    }
}

extern "C" void kernel_launch(void* const* d_in, const int* in_sizes, int n_in,
                              void* d_out, int out_size, void* d_ws, size_t ws_size,
                              hipStream_t stream) {
    const float* x = (const float*)d_in[0];
    const float* w = (const float*)d_in[1];
    float*     out = (float*)d_out;
    const int Btotal = in_sizes[0] / 196;        // 14*14*1; 131072 -> multiple of BT
    const int blocks = Btotal / BT;
    hipLaunchKernelGGL(lc_conv2d_wmma_f32, dim3(blocks), dim3(THREADS), 0, stream,
                       x, w, out, Btotal);
}